// OverAll_6734508720516
// MI455X (gfx1250) — compile-verified
//
#include <hip/hip_runtime.h>

#define N_NODES 30000
#define DIM     100
#define M_EDGES 250000
#define R_REL   1000
#define T_TIME  500
#define EE_ENT  500000
#define EE_REL  60000
#define EE_TIME 60000
#define OUTW    600

typedef __attribute__((ext_vector_type(2))) float v2f;
typedef __attribute__((ext_vector_type(8))) float v8f;

// ---- ordered-uint encoding for float atomic max ----
__device__ __forceinline__ unsigned fenc(float f) {
  unsigned u = __float_as_uint(f);
  return (u & 0x80000000u) ? ~u : (u | 0x80000000u);
}
__device__ __forceinline__ float fdec(unsigned e) {
  unsigned u = (e & 0x80000000u) ? (e & 0x7fffffffu) : ~e;
  return __uint_as_float(u);
}

// ---------------- elementwise / scatter kernels ----------------
__global__ void fill_k(float* p, float v, int n) {
  int i = blockIdx.x * blockDim.x + threadIdx.x;
  if (i < n) p[i] = v;
}

__global__ void degree_k(const int* __restrict__ idx, float* __restrict__ deg, int E) {
  int e = blockIdx.x * blockDim.x + threadIdx.x;
  if (e < E) atomicAdd(&deg[idx[2 * e]], 1.0f);
}

// wave-per-edge scatter-add of embedding rows
__global__ void scatter_k(const int* __restrict__ idx, const float* __restrict__ emb,
                          float* __restrict__ feat, int E) {
  int w = blockIdx.x * (blockDim.x >> 5) + (threadIdx.x >> 5);
  int lane = threadIdx.x & 31;
  if (w >= E) return;
  int row = idx[2 * w], col = idx[2 * w + 1];
  const float* s = emb + (size_t)col * DIM;
  float* d = feat + (size_t)row * DIM;
  for (int j = lane; j < DIM; j += 32) atomicAdd(&d[j], s[j]);
}

__global__ void divdeg_k(float* __restrict__ feat, const float* __restrict__ deg, int n) {
  int i = blockIdx.x * blockDim.x + threadIdx.x;
  if (i < n) feat[i] /= fmaxf(deg[i / DIM], 1.0f);
}

// relu(src) -> feats buffer AND strided column slice of output
__global__ void reluout_k(const float* __restrict__ src, float* __restrict__ dst,
                          float* __restrict__ out, int coloff) {
  int i = blockIdx.x * blockDim.x + threadIdx.x;
  if (i >= N_NODES * DIM) return;
  float v = fmaxf(src[i], 0.0f);
  dst[i] = v;
  int r = i / DIM, d = i - r * DIM;
  out[(size_t)r * OUTW + coloff + d] = v;
}

// L1 norm of each embedding row (wave per row)
__global__ void l1_rows_k(const float* __restrict__ emb, float* __restrict__ L1, int nrows) {
  int r = blockIdx.x * (blockDim.x >> 5) + (threadIdx.x >> 5);
  int lane = threadIdx.x & 31;
  if (r >= nrows) return;
  float s = 0.0f;
  for (int d = lane; d < DIM; d += 32) s += fabsf(emb[(size_t)r * DIM + d]);
  for (int o = 16; o; o >>= 1) s += __shfl_xor(s, o, 32);
  if (lane == 0) L1[r] = s;
}

// rscale[e] = r_val[e] / max(|r_val[e]| * L1(emb[rid]), 1e-12)
__global__ void rscale_k(const float* __restrict__ rv, const int* __restrict__ ridx,
                         const float* __restrict__ L1, float* __restrict__ rsc) {
  int e = blockIdx.x * blockDim.x + threadIdx.x;
  if (e >= M_EDGES) return;
  float v = rv[e];
  float l = fabsf(v) * L1[ridx[2 * e + 1]];
  rsc[e] = v / fmaxf(l, 1e-12f);
}

// ---------------- WMMA kernels (v_wmma_f32_16x16x4_f32) ----------------
// dot of 16 matrix rows with 2 kernel vectors: out[r*2+{0,1}] = row·k0, row·k1
__global__ void dotk_wmma_k(const float* __restrict__ mat, int nrows,
                            const float* __restrict__ k0, const float* __restrict__ k1,
                            float* __restrict__ outd) {
  __shared__ float tile[8][16][17];
  int w8 = threadIdx.x >> 5;
  int tileId = blockIdx.x * 8 + w8;
  int lane = threadIdx.x & 31;
  int ntiles = (nrows + 15) / 16;
  if (tileId >= ntiles) return;
  int eL = lane & 15;
  int r = tileId * 16 + eL;
  int rr = (r < nrows) ? r : (nrows - 1);
  const float* mp = mat + (size_t)rr * DIM;
  const float* kp = (eL == 1) ? k1 : k0;
  float km = (eL < 2) ? 1.0f : 0.0f;
  int off = (lane & 16) ? 2 : 0;
  v8f C = {};
#pragma unroll
  for (int c = 0; c < 25; ++c) {
    int o = 4 * c + off;
    v2f a; a.x = mp[o];        a.y = mp[o + 1];
    v2f b; b.x = km * kp[o];   b.y = km * kp[o + 1];
    C = __builtin_amdgcn_wmma_f32_16x16x4_f32(false, a, false, b, (short)0, C, false, false);
  }
  int mB = (lane & 16) ? 8 : 0;
#pragma unroll
  for (int q = 0; q < 8; ++q) tile[w8][mB + q][eL] = C[q];
  __builtin_amdgcn_wave_barrier();
  if (lane < 16 && r < nrows) {
    outd[2 * r]     = tile[w8][eL][0];
    outd[2 * r + 1] = tile[w8][eL][1];
  }
}

// per-edge pairwise dot sdot[e] = feats[col_e] · emb[rid_e]  (diag of 16x16 WMMA tile)
__global__ void pairdot_wmma_k(const float* __restrict__ feats, const float* __restrict__ emb,
                               const int* __restrict__ adj, const int* __restrict__ ridx,
                               float* __restrict__ sdot) {
  __shared__ float tile[8][16][17];
  int w8 = threadIdx.x >> 5;
  int tileId = blockIdx.x * 8 + w8;
  int lane = threadIdx.x & 31;
  const int ntiles = M_EDGES / 16;
  if (tileId >= ntiles) return;
  int eL = lane & 15;
  int e = tileId * 16 + eL;
  int col = adj[2 * e + 1];
  int rid = ridx[2 * e + 1];
  const float* fp = feats + (size_t)col * DIM;
  const float* rp = emb + (size_t)rid * DIM;
  int off = (lane & 16) ? 2 : 0;
  v8f C = {};
#pragma unroll
  for (int c = 0; c < 25; ++c) {
    int o = 4 * c + off;
    v2f a; a.x = fp[o]; a.y = fp[o + 1];
    v2f b; b.x = rp[o]; b.y = rp[o + 1];
    C = __builtin_amdgcn_wmma_f32_16x16x4_f32(false, a, false, b, (short)0, C, false, false);
  }
  int mB = (lane & 16) ? 8 : 0;
#pragma unroll
  for (int q = 0; q < 8; ++q) tile[w8][mB + q][eL] = C[q];
  __builtin_amdgcn_wave_barrier();
  if (lane < 16) sdot[e] = tile[w8][eL][eL];
}

// ---------------- attention kernels ----------------
__global__ void att_k(const int* __restrict__ adj, const int* __restrict__ ridx,
                      const float* __restrict__ nd, const float* __restrict__ ed,
                      const float* __restrict__ rsc, const float* __restrict__ sdot,
                      float* __restrict__ att1, unsigned* __restrict__ mEnc) {
  int e = blockIdx.x * blockDim.x + threadIdx.x;
  if (e >= M_EDGES) return;
  int row = adj[2 * e], col = adj[2 * e + 1], rid = ridx[2 * e + 1];
  float rs = rsc[e];
  float s = rs * sdot[e];
  float a = nd[2 * row] + nd[2 * col + 1] - 2.0f * s * rs * ed[2 * rid] + rs * ed[2 * rid + 1];
  att1[e] = a;
  atomicMax(&mEnc[row], fenc(a));
}

__global__ void expden_k(const int* __restrict__ adj, const float* __restrict__ att1,
                         const unsigned* __restrict__ mEnc, float* __restrict__ eatt,
                         float* __restrict__ denom) {
  int e = blockIdx.x * blockDim.x + threadIdx.x;
  if (e >= M_EDGES) return;
  int row = adj[2 * e];
  float ev = expf(att1[e] - fdec(mEnc[row]));
  eatt[e] = ev;
  atomicAdd(&denom[row], ev);
}

// agg[row] += w * (feats[col] - 2*s*rscale*emb[rid]), wave per edge
__global__ void agg_k(const int* __restrict__ adj, const int* __restrict__ ridx,
                      const float* __restrict__ feats, const float* __restrict__ emb,
                      const float* __restrict__ rsc, const float* __restrict__ sdot,
                      const float* __restrict__ eatt, const float* __restrict__ denom,
                      float* __restrict__ agg) {
  int w = blockIdx.x * (blockDim.x >> 5) + (threadIdx.x >> 5);
  int lane = threadIdx.x & 31;
  if (w >= M_EDGES) return;
  int row = adj[2 * w], col = adj[2 * w + 1], rid = ridx[2 * w + 1];
  float rs = rsc[w];
  float coef = 2.0f * rs * rs * sdot[w];
  float wt = eatt[w] / denom[row];
  const float* fp = feats + (size_t)col * DIM;
  const float* rp = emb + (size_t)rid * DIM;
  float* ap = agg + (size_t)row * DIM;
  for (int d = lane; d < DIM; d += 32) atomicAdd(&ap[d], wt * (fp[d] - coef * rp[d]));
}

// ---------------- host orchestration ----------------
static void run_ga(const float* feature, const float* EMB, int embRows,
                   const int* adj, const int* ridx, const float* rsc,
                   const float* Kern, float* outbase, int coloff,
                   float* bufA, float* bufB, float* nd, float* ed,
                   unsigned* mEnc, float* denom, float* sdot, float* att1, float* eatt,
                   hipStream_t st) {
  const int ND = N_NODES * DIM;
  const int bND = (ND + 255) / 256;
  const int bN = (N_NODES + 255) / 256;
  const int bM = (M_EDGES + 255) / 256;
  reluout_k<<<bND, 256, 0, st>>>(feature, bufA, outbase, coloff);
  for (int l = 0; l < 2; ++l) {
    const float* K = Kern + l * 300;
    dotk_wmma_k<<<(((N_NODES + 15) / 16) + 7) / 8, 256, 0, st>>>(bufA, N_NODES, K, K + 100, nd);
    dotk_wmma_k<<<(((embRows + 15) / 16) + 7) / 8, 256, 0, st>>>(EMB, embRows, K + 100, K + 200, ed);
    pairdot_wmma_k<<<((M_EDGES / 16) + 7) / 8, 256, 0, st>>>(bufA, EMB, adj, ridx, sdot);
    fill_k<<<bN, 256, 0, st>>>((float*)mEnc, 0.0f, N_NODES);
    fill_k<<<bN, 256, 0, st>>>(denom, 0.0f, N_NODES);
    att_k<<<bM, 256, 0, st>>>(adj, ridx, nd, ed, rsc, sdot, att1, mEnc);
    expden_k<<<bM, 256, 0, st>>>(adj, att1, mEnc, eatt, denom);
    fill_k<<<bND, 256, 0, st>>>(bufB, 0.0f, ND);
    agg_k<<<(M_EDGES + 7) / 8, 256, 0, st>>>(adj, ridx, bufA, EMB, rsc, sdot, eatt, denom, bufB);
    reluout_k<<<bND, 256, 0, st>>>(bufB, bufA, outbase, coloff + (l + 1) * DIM);
  }
}

extern "C" void kernel_launch(void* const* d_in, const int* in_sizes, int n_in,
                              void* d_out, int out_size, void* d_ws, size_t ws_size,
                              hipStream_t stream) {
  const int*   adj      = (const int*)d_in[0];
  const int*   rIdx     = (const int*)d_in[1];
  const float* rVal     = (const float*)d_in[2];
  const int*   tIdx     = (const int*)d_in[3];
  const int*   entAdj   = (const int*)d_in[4];
  const int*   relAdj   = (const int*)d_in[5];
  const int*   timeAdj  = (const int*)d_in[6];
  const float* entEmbR  = (const float*)d_in[7];
  const float* entEmbT  = (const float*)d_in[8];
  const float* relEmb   = (const float*)d_in[9];
  const float* timeEmb  = (const float*)d_in[10];
  const float* eK       = (const float*)d_in[11];
  const float* tK       = (const float*)d_in[12];
  float* out = (float*)d_out;

  // workspace carve (floats)
  float* W = (float*)d_ws;
  size_t off = 0;
  auto take = [&](size_t n) { float* p = W + off; off += n; return p; };
  float* feat_r  = take((size_t)N_NODES * DIM);
  float* feat_t  = take((size_t)N_NODES * DIM);
  float* relF    = take((size_t)N_NODES * DIM);
  float* timeF   = take((size_t)N_NODES * DIM);
  float* bufA    = take((size_t)N_NODES * DIM);
  float* bufB    = take((size_t)N_NODES * DIM);
  float* degE    = take(N_NODES);
  float* degR    = take(N_NODES);
  float* degT    = take(N_NODES);
  unsigned* mEnc = (unsigned*)take(N_NODES);
  float* denom   = take(N_NODES);
  float* nd      = take((size_t)2 * N_NODES);
  float* L1tab   = take(R_REL);
  float* ed      = take((size_t)2 * R_REL);
  float* rsc     = take(M_EDGES);
  float* sdot    = take(M_EDGES);
  float* att1    = take(M_EDGES);
  float* eatt    = take(M_EDGES);

  const int ND = N_NODES * DIM;
  const int bND = (ND + 255) / 256;
  const int bN = (N_NODES + 255) / 256;
  const int bM = (M_EDGES + 255) / 256;

  // ---- norm_adj_matmul features ----
  fill_k<<<bN, 256, 0, stream>>>(degE, 0.0f, N_NODES);
  fill_k<<<bN, 256, 0, stream>>>(degR, 0.0f, N_NODES);
  fill_k<<<bN, 256, 0, stream>>>(degT, 0.0f, N_NODES);
  degree_k<<<(EE_ENT + 255) / 256, 256, 0, stream>>>(entAdj, degE, EE_ENT);
  degree_k<<<(EE_REL + 255) / 256, 256, 0, stream>>>(relAdj, degR, EE_REL);
  degree_k<<<(EE_TIME + 255) / 256, 256, 0, stream>>>(timeAdj, degT, EE_TIME);

  fill_k<<<bND, 256, 0, stream>>>(feat_r, 0.0f, ND);
  scatter_k<<<(EE_ENT + 7) / 8, 256, 0, stream>>>(entAdj, entEmbR, feat_r, EE_ENT);
  divdeg_k<<<bND, 256, 0, stream>>>(feat_r, degE, ND);

  fill_k<<<bND, 256, 0, stream>>>(feat_t, 0.0f, ND);
  scatter_k<<<(EE_ENT + 7) / 8, 256, 0, stream>>>(entAdj, entEmbT, feat_t, EE_ENT);
  divdeg_k<<<bND, 256, 0, stream>>>(feat_t, degE, ND);

  fill_k<<<bND, 256, 0, stream>>>(relF, 0.0f, ND);
  scatter_k<<<(EE_REL + 7) / 8, 256, 0, stream>>>(relAdj, relEmb, relF, EE_REL);
  divdeg_k<<<bND, 256, 0, stream>>>(relF, degR, ND);

  fill_k<<<bND, 256, 0, stream>>>(timeF, 0.0f, ND);
  scatter_k<<<(EE_TIME + 7) / 8, 256, 0, stream>>>(timeAdj, timeEmb, timeF, EE_TIME);
  divdeg_k<<<bND, 256, 0, stream>>>(timeF, degT, ND);

  // ---- relation pair (rel_emb, r_index, e_kernels) ----
  l1_rows_k<<<(R_REL + 7) / 8, 256, 0, stream>>>(relEmb, L1tab, R_REL);
  rscale_k<<<bM, 256, 0, stream>>>(rVal, rIdx, L1tab, rsc);
  run_ga(feat_r, relEmb, R_REL, adj, rIdx, rsc, eK, out, 0,
         bufA, bufB, nd, ed, mEnc, denom, sdot, att1, eatt, stream);
  run_ga(relF, relEmb, R_REL, adj, rIdx, rsc, eK, out, 300,
         bufA, bufB, nd, ed, mEnc, denom, sdot, att1, eatt, stream);

  // ---- time pair (time_emb, t_index, t_kernels) ----
  l1_rows_k<<<(T_TIME + 7) / 8, 256, 0, stream>>>(timeEmb, L1tab, T_TIME);
  rscale_k<<<bM, 256, 0, stream>>>(rVal, tIdx, L1tab, rsc);
  float* out2 = out + (size_t)N_NODES * OUTW;
  run_ga(feat_t, timeEmb, T_TIME, adj, tIdx, rsc, tK, out2, 0,
         bufA, bufB, nd, ed, mEnc, denom, sdot, att1, eatt, stream);
  run_ga(timeF, timeEmb, T_TIME, adj, tIdx, rsc, tK, out2, 300,
         bufA, bufB, nd, ed, mEnc, denom, sdot, att1, eatt, stream);
}